// gcn_59811714564162
// MI455X (gfx1250) — compile-verified
//
#include <hip/hip_runtime.h>
#include <hip/hip_bf16.h>

// Problem constants
#define BB 64
#define CC 32
#define NN 1000
#define LL 12
#define CL (CC * LL)           // 384
#define NL (NN * LL)           // 12000
#define BSTRIDE (CC * NL)      // 384000 floats per batch (x, t, and out all [*,32,1000,12])
#define WROW 224               // 7*C
#define MTILES 63              // ceil(1000/16)

typedef __attribute__((ext_vector_type(2))) float v2f;
typedef __attribute__((ext_vector_type(8))) float v8f;

// ---------------------------------------------------------------------------
// Kernel 0: out[b,o,n,l] = bias[o] + sum_c W[o,c] * x[b,c,n,l]   (piece k=0)
// Plain-store initializes d_out for the += epilogues of the WMMA kernels.
// ---------------------------------------------------------------------------
__global__ __launch_bounds__(256) void init_mix(const float* __restrict__ X,
                                                const float* __restrict__ W,
                                                const float* __restrict__ bias,
                                                float* __restrict__ out) {
  __shared__ float lds[CC * 16 * LL];  // 24 KB: [c][m][l]
  const int tid = threadIdx.x;
  const int mtile = blockIdx.x;
  const int b = blockIdx.y;
  const float* xb = X + (size_t)b * BSTRIDE;

  for (int idx = tid; idx < CC * 16 * LL; idx += 256) {
    int c = idx / (16 * LL);
    int r = idx % (16 * LL);
    int m = r / LL;
    int l = r % LL;
    int mg = mtile * 16 + m;
    lds[idx] = (mg < NN) ? xb[c * NL + mg * LL + l] : 0.0f;
  }
  __syncthreads();

  for (int idx = tid; idx < CC * 16 * LL; idx += 256) {
    int o = idx / (16 * LL);
    int r = idx % (16 * LL);
    int m = r / LL;
    int l = r % LL;
    int mg = mtile * 16 + m;
    if (mg < NN) {
      float s = bias[o];
#pragma unroll
      for (int c = 0; c < CC; ++c) s += W[o * WROW + c] * lds[(c * 16 + m) * LL + l];
      out[(size_t)b * BSTRIDE + o * NL + mg * LL + l] = s;
    }
  }
}

// ---------------------------------------------------------------------------
// WMMA diffusion kernel: Y[b,c,m,l] = sum_n A[n,m] * X[b,c,n,l]
// Fused epilogue: out[b,o,m,l] += sum_c W[o, w_block*32 + c] * Y[b,c,m,l]
// One workgroup = (b, 16-wide m tile). 8 waves x 3 col-tiles cover 384 (c,l) cols.
// K loop: 250 steps of v_wmma_f32_16x16x4_f32 per output tile.
// ---------------------------------------------------------------------------
__global__ __launch_bounds__(256) void diffuse_mix(const float* __restrict__ X,
                                                   const float* __restrict__ A,
                                                   float* __restrict__ Y,
                                                   const float* __restrict__ W,
                                                   float* __restrict__ out,
                                                   int w_block, int write_y) {
  __shared__ float lds[CC * 16 * LL];  // 24 KB result staging: [c][m][l]
  const int tid  = threadIdx.x;
  const int wave = tid >> 5;
  const int lane = tid & 31;
  const int half = lane >> 4;   // 0: lanes 0-15, 1: lanes 16-31
  const int l16  = lane & 15;
  const int mtile = blockIdx.x;
  const int b = blockIdx.y;

  // A operand: Aop[m=l16, k=2*half+j] = A[(n0 + 2*half + j)*NN + mg]
  const int mg_a = (mtile * 16 + l16 < NN) ? (mtile * 16 + l16) : (NN - 1);
  const float* aptr = A + (size_t)(2 * half) * NN + mg_a;

  // B operand per col-tile: Bop[k=2*half+j, col=l16] = X[b, c, n0+2*half+j, l]
  const float* xb = X + (size_t)b * BSTRIDE;
  int colg0 = (wave * 3 + 0) * 16 + l16;
  int colg1 = (wave * 3 + 1) * 16 + l16;
  int colg2 = (wave * 3 + 2) * 16 + l16;
  const float* xp0 = xb + (colg0 / LL) * NL + (colg0 % LL) + (size_t)(2 * half) * LL;
  const float* xp1 = xb + (colg1 / LL) * NL + (colg1 % LL) + (size_t)(2 * half) * LL;
  const float* xp2 = xb + (colg2 / LL) * NL + (colg2 % LL) + (size_t)(2 * half) * LL;

  v8f acc0 = {}, acc1 = {}, acc2 = {};

  for (int n0 = 0; n0 < NN; n0 += 4) {
    // prefetch A four k-steps ahead (emits global_prefetch_b8)
    __builtin_prefetch(aptr + 16 * NN, 0, 1);

    v2f av;
    av.x = aptr[0];
    av.y = aptr[NN];
    aptr += 4 * NN;

    v2f bv0, bv1, bv2;
    bv0.x = xp0[0]; bv0.y = xp0[LL];
    bv1.x = xp1[0]; bv1.y = xp1[LL];
    bv2.x = xp2[0]; bv2.y = xp2[LL];
    xp0 += 4 * LL; xp1 += 4 * LL; xp2 += 4 * LL;

    acc0 = __builtin_amdgcn_wmma_f32_16x16x4_f32(false, av, false, bv0, (short)0, acc0, false, false);
    acc1 = __builtin_amdgcn_wmma_f32_16x16x4_f32(false, av, false, bv1, (short)0, acc1, false, false);
    acc2 = __builtin_amdgcn_wmma_f32_16x16x4_f32(false, av, false, bv2, (short)0, acc2, false, false);
  }

  // Scatter D tiles into LDS [c][m][l] and (odd-order pieces only) to global Y.
  {
    v8f accs[3] = {acc0, acc1, acc2};
    int colgs[3] = {colg0, colg1, colg2};
#pragma unroll
    for (int t = 0; t < 3; ++t) {
      int c = colgs[t] / LL;
      int l = colgs[t] % LL;
#pragma unroll
      for (int i = 0; i < 8; ++i) {
        int m = i + 8 * half;          // C/D layout: vgpr i -> rows i, i+8
        float v = accs[t][i];
        lds[(c * 16 + m) * LL + l] = v;
        if (write_y) {
          int mg = mtile * 16 + m;
          if (mg < NN) Y[(size_t)b * BSTRIDE + c * NL + mg * LL + l] = v;
        }
      }
    }
  }
  __syncthreads();

  // Channel mix with W block k, accumulate into out (unique writer per element,
  // kernels are stream-serialized so plain RMW is safe).
  const float* Wblk = W + w_block * CC;
  for (int idx = tid; idx < CC * 16 * LL; idx += 256) {
    int o = idx / (16 * LL);
    int r = idx % (16 * LL);
    int m = r / LL;
    int l = r % LL;
    int mg = mtile * 16 + m;
    if (mg < NN) {
      float s = 0.0f;
#pragma unroll
      for (int c = 0; c < CC; ++c) s += Wblk[o * WROW + c] * lds[(c * 16 + m) * LL + l];
      out[(size_t)b * BSTRIDE + o * NL + mg * LL + l] += s;
    }
  }
}

// ---------------------------------------------------------------------------
// Launch: out = W0 x + b, then for each A: t = A^T x (mix W1), A^T t (mix W2, no store)
// Workspace: single ping buffer (B*C*N*L floats = 98.3 MB) in d_ws.
// ---------------------------------------------------------------------------
extern "C" void kernel_launch(void* const* d_in, const int* in_sizes, int n_in,
                              void* d_out, int out_size, void* d_ws, size_t ws_size,
                              hipStream_t stream) {
  const float* x  = (const float*)d_in[0];
  const float* a0 = (const float*)d_in[1];
  const float* a1 = (const float*)d_in[2];
  const float* a2 = (const float*)d_in[3];
  const float* W  = (const float*)d_in[4];
  const float* bi = (const float*)d_in[5];
  float* out = (float*)d_out;
  float* t = (float*)d_ws;  // needs BB*CC*NN*LL*4 = 98,304,000 bytes

  dim3 grid(MTILES, BB);
  dim3 block(256);

  init_mix<<<grid, block, 0, stream>>>(x, W, bi, out);

  // a0: first order (store t), second order (mix only)
  diffuse_mix<<<grid, block, 0, stream>>>(x, a0, t, W, out, 1, 1);
  diffuse_mix<<<grid, block, 0, stream>>>(t, a0, t, W, out, 2, 0);
  // a1
  diffuse_mix<<<grid, block, 0, stream>>>(x, a1, t, W, out, 3, 1);
  diffuse_mix<<<grid, block, 0, stream>>>(t, a1, t, W, out, 4, 0);
  // a2
  diffuse_mix<<<grid, block, 0, stream>>>(x, a2, t, W, out, 5, 1);
  diffuse_mix<<<grid, block, 0, stream>>>(t, a2, t, W, out, 6, 0);
}